// Ranking_35493609734887
// MI455X (gfx1250) — compile-verified
//
#include <hip/hip_runtime.h>
#include <hip/hip_bf16.h>
#include <stdint.h>

#define BS      4096
#define SEGLEN  8192
#define LAM1    8e-05f
#define LAM2    8e-05f

typedef float v2f __attribute__((ext_vector_type(2)));
typedef float v8f __attribute__((ext_vector_type(8)));

// ---------------------------------------------------------------------------
// Kernel 1: per-row reduction. One block per row (256 threads = 8 wave32).
// Stage the 32KB row into LDS with async B128 loads, then compute
//   sum, sum((s[j+1]-s[j])^2), argmax-index (first occurrence)
// Sums are accumulated through V_WMMA_F32_16X16X4_F32 against an all-ones B.
// ---------------------------------------------------------------------------
__global__ __launch_bounds__(256) void row_reduce_kernel(
    const float* __restrict__ scores,
    float* __restrict__ osum,
    float* __restrict__ osmooth,
    int*   __restrict__ oms)
{
    __shared__ __align__(16) float rowbuf[SEGLEN];
    __shared__ float wsum[8], wsm[8], wbv[8];
    __shared__ int   wbi[8];

    const int tid  = threadIdx.x;
    const int row  = blockIdx.x;
    const int lane = tid & 31;
    const int wv   = tid >> 5;

    const char*    gbase   = (const char*)(scores + (size_t)row * SEGLEN);
    const uint32_t ldsbase = (uint32_t)(uintptr_t)rowbuf;   // group-relative LDS offset

    // ---- async stage: 256 threads x 8 issues x 16B = 32KB --------------------
#pragma unroll
    for (int r = 0; r < 8; ++r) {
        const int      byteoff = r * 4096 + tid * 16;
        const uint32_t lds     = ldsbase + (uint32_t)byteoff;
        const char*    g       = gbase + byteoff;
        asm volatile("global_load_async_to_lds_b128 %0, %1, off"
                     :: "v"(lds), "v"(g) : "memory");
    }
    asm volatile("s_wait_asynccnt 0" ::: "memory");
    __syncthreads();

    // ---- per-wave compute over its 1024-element slice ------------------------
    v8f accS = {0.f,0.f,0.f,0.f,0.f,0.f,0.f,0.f};   // running row sum
    v8f accD = {0.f,0.f,0.f,0.f,0.f,0.f,0.f,0.f};   // running sum of d^2
    const v2f bOnes = {1.0f, 1.0f};

    float bestv = -3.4e38f;
    int   besti = 0;

#pragma unroll 4
    for (int c = 0; c < 16; ++c) {
        const int i0 = wv * 1024 + c * 64 + lane * 2;   // two consecutive elems/lane
        const float s0 = rowbuf[i0];
        const float s1 = rowbuf[i0 + 1];
        int i2 = i0 + 2;
        if (i2 > SEGLEN - 1) i2 = SEGLEN - 1;           // clamp (one lane only)
        const float s2 = rowbuf[i2];

        const float dx = s1 - s0;                                 // diff j = i0 (always valid)
        const float dy = (i0 + 1 < SEGLEN - 1) ? (s2 - s1) : 0.f; // diff j = i0+1

        const v2f a  = {s0, s1};
        const v2f dd = {dx * dx, dy * dy};

        // D[m,n] = sum_k A[m,k] + C[m,n]  (all-ones B): matrix-pipe accumulate
        accS = __builtin_amdgcn_wmma_f32_16x16x4_f32(
                   false, a,  false, bOnes, (short)0, accS, false, false);
        accD = __builtin_amdgcn_wmma_f32_16x16x4_f32(
                   false, dd, false, bOnes, (short)0, accD, false, false);

        // argmax, first-occurrence (ascending index within lane, strict >)
        if (s0 > bestv) { bestv = s0; besti = i0; }
        if (s1 > bestv) { bestv = s1; besti = i0 + 1; }
    }

    // All 16 columns of D are identical -> sum over all lanes/VGPRs = 16*total.
    float tS = accS[0]+accS[1]+accS[2]+accS[3]+accS[4]+accS[5]+accS[6]+accS[7];
    float tD = accD[0]+accD[1]+accD[2]+accD[3]+accD[4]+accD[5]+accD[6]+accD[7];

    for (int off = 16; off > 0; off >>= 1) {
        tS += __shfl_down(tS, off);
        tD += __shfl_down(tD, off);
        const float ov = __shfl_down(bestv, off);
        const int   oi = __shfl_down(besti, off);
        if (ov > bestv || (ov == bestv && oi < besti)) { bestv = ov; besti = oi; }
    }

    if (lane == 0) {
        wsum[wv] = tS * 0.0625f;   // exact /16 correction
        wsm[wv]  = tD * 0.0625f;
        wbv[wv]  = bestv;
        wbi[wv]  = besti;
    }
    __syncthreads();

    if (tid == 0) {
        float S = 0.f, D = 0.f;
        float bv = wbv[0]; int bi = wbi[0];
        for (int k = 0; k < 8; ++k) {
            S += wsum[k]; D += wsm[k];
            if (wbv[k] > bv || (wbv[k] == bv && wbi[k] < bi)) { bv = wbv[k]; bi = wbi[k]; }
        }
        osum[row]    = S;
        osmooth[row] = D;
        oms[row]     = bi;
    }
}

// ---------------------------------------------------------------------------
// Kernel 2: per-row margin over negative columns + per-row loss term.
// margin_i = sum_{j : labels[j]==0} max(0, 1 - ms_i + ms_j)   (int32, exact)
// per_i    = labels[i]==1 ? LAM1*smooth + LAM2*sum + float(margin) : 0
// ---------------------------------------------------------------------------
__global__ __launch_bounds__(256) void margin_kernel(
    const int*   __restrict__ labels,
    const float* __restrict__ osum,
    const float* __restrict__ osmooth,
    const int*   __restrict__ oms,
    float*       __restrict__ per)
{
    __shared__ int wacc[8];
    const int i   = blockIdx.x;
    const int tid = threadIdx.x;
    const int msi = oms[i];

    int acc = 0;
    for (int j = tid; j < BS; j += 256) {
        const int isneg = (labels[j] == 0) ? 1 : 0;
        int v = 1 - msi + oms[j];
        v = (v > 0) ? v : 0;
        acc += v * isneg;
    }
    for (int off = 16; off > 0; off >>= 1) acc += __shfl_down(acc, off);
    if ((tid & 31) == 0) wacc[tid >> 5] = acc;
    __syncthreads();

    if (tid == 0) {
        int m = 0;
        for (int k = 0; k < 8; ++k) m += wacc[k];
        per[i] = (labels[i] == 1)
                     ? (LAM1 * osmooth[i] + LAM2 * osum[i] + (float)m)
                     : 0.0f;
    }
}

// ---------------------------------------------------------------------------
// Kernel 3: deterministic final mean over 4096 per-row terms.
// ---------------------------------------------------------------------------
__global__ __launch_bounds__(256) void finalize_kernel(
    const float* __restrict__ per, float* __restrict__ out)
{
    __shared__ float wacc[8];
    const int tid = threadIdx.x;
    float acc = 0.f;
    for (int j = tid; j < BS; j += 256) acc += per[j];   // fixed order
    for (int off = 16; off > 0; off >>= 1) acc += __shfl_down(acc, off);
    if ((tid & 31) == 0) wacc[tid >> 5] = acc;
    __syncthreads();
    if (tid == 0) {
        float s = 0.f;
        for (int k = 0; k < 8; ++k) s += wacc[k];
        out[0] = s * (1.0f / (float)BS);
    }
}

extern "C" void kernel_launch(void* const* d_in, const int* in_sizes, int n_in,
                              void* d_out, int out_size, void* d_ws, size_t ws_size,
                              hipStream_t stream)
{
    const float* scores = (const float*)d_in[0];
    const int*   labels = (const int*)d_in[1];

    float* ws      = (float*)d_ws;
    float* osum    = ws;                 // [BS]
    float* osmooth = ws + BS;            // [BS]
    int*   oms     = (int*)(ws + 2*BS);  // [BS]
    float* per     = ws + 3*BS;          // [BS]

    row_reduce_kernel<<<BS, 256, 0, stream>>>(scores, osum, osmooth, oms);
    margin_kernel   <<<BS, 256, 0, stream>>>(labels, osum, osmooth, oms, per);
    finalize_kernel <<<1,  256, 0, stream>>>(per, (float*)d_out);
}